// MAg_90709709292194
// MI455X (gfx1250) — compile-verified
//
#include <hip/hip_runtime.h>

#define NN 2048          // nodes
#define NCH 4            // output channels == input channels
#define XROW 8192        // inputs row stride (n * in_chan)

typedef __attribute__((ext_vector_type(2))) float v2f;
typedef __attribute__((ext_vector_type(8))) float v8f;

// K1: wave per row -> row_nnz[r] and dense local-index matrix lidx[r][c]
__global__ void __launch_bounds__(256) k_row_scan(const float* __restrict__ adj,
                                                  int* __restrict__ row_nnz,
                                                  unsigned short* __restrict__ lidx) {
    const int lane = threadIdx.x & 31;
    const int wave = threadIdx.x >> 5;
    const int r = blockIdx.x * 8 + wave;
    const float* rowp = adj + (size_t)r * NN;
    unsigned short* lp = lidx + (size_t)r * NN;
    const unsigned lmask = (1u << lane) - 1u;
    unsigned run = 0;
    for (int base = 0; base < NN; base += 32) {
        float v = rowp[base + lane];
        unsigned m = __builtin_amdgcn_ballot_w32(v != 0.0f);
        lp[base + lane] = (unsigned short)(run + (unsigned)__builtin_popcount(m & lmask));
        run += (unsigned)__builtin_popcount(m);
    }
    if (lane == 0) row_nnz[r] = (int)run;
}

// K2: thread per column -> col_nnz[c] (coalesced across lanes for each r)
__global__ void k_col_count(const float* __restrict__ adj, int* __restrict__ col_nnz) {
    const int c = blockIdx.x * blockDim.x + threadIdx.x;
    if (c >= NN) return;
    int cnt = 0;
    for (int r = 0; r < NN; ++r)
        cnt += (adj[(size_t)r * NN + c] != 0.0f) ? 1 : 0;
    col_nnz[c] = cnt;
}

// K3: exclusive scan of 2048 ints -> out[0..2048] (out[2048] = total)
__global__ void __launch_bounds__(1024) k_scan2048(const int* __restrict__ in,
                                                   int* __restrict__ out) {
    __shared__ int lds[1024];
    const int t = threadIdx.x;
    const int x0 = in[2 * t], x1 = in[2 * t + 1];
    const int pair = x0 + x1;
    lds[t] = pair;
    __syncthreads();
    for (int off = 1; off < 1024; off <<= 1) {
        int add = (t >= off) ? lds[t - off] : 0;
        __syncthreads();
        lds[t] += add;
        __syncthreads();
    }
    const int incl = lds[t];
    const int excl = incl - pair;
    out[2 * t] = excl;
    out[2 * t + 1] = excl + x0;
    if (t == 1023) out[2048] = incl;
}

// K4: thread per column -> CSC entries in deterministic (r ascending) order
__global__ void k_csc_build(const float* __restrict__ adj, const int* __restrict__ rowptr,
                            const int* __restrict__ row_nnz,
                            const unsigned short* __restrict__ lidx,
                            const int* __restrict__ colptr,
                            int* __restrict__ e_r, int* __restrict__ e_wb,
                            int* __restrict__ e_rn) {
    const int c = blockIdx.x * blockDim.x + threadIdx.x;
    if (c >= NN) return;
    int wp = colptr[c];
    for (int r = 0; r < NN; ++r) {
        if (adj[(size_t)r * NN + c] != 0.0f) {
            e_r[wp]  = r;
            e_wb[wp] = 4 * rowptr[r] + (int)lidx[(size_t)r * NN + c];
            e_rn[wp] = row_nnz[r];
            ++wp;
        }
    }
}

// K5: wave per output column c. Per edge: D(32x4) += X_r(32x4) @ W_e(4x4)
// via two V_WMMA_F32_16X16X4_F32 (batch halves), K = 4 input channels,
// N lanes 0..3 = output channels (bias pre-loaded into accumulator).
// Inner loop is branch-free: weight index uses clamped N (provably in-bounds:
// max = 3*4nnz + 4*rowptr(r) + 3*rn + (rn-1) <= 16*nnz - 1); lanes N=4..15
// accumulate finite junk that is never stored.
__global__ void __launch_bounds__(256) k_spmm_wmma(
    const float* __restrict__ x, const float* __restrict__ kw,
    const float* __restrict__ bias, float* __restrict__ out,
    const int* __restrict__ colptr, const int* __restrict__ rowptr,
    const int* __restrict__ e_r, const int* __restrict__ e_wb,
    const int* __restrict__ e_rn) {
    const int lane = threadIdx.x & 31;
    const int wave = __builtin_amdgcn_readfirstlane((int)(threadIdx.x >> 5));
    const int c    = blockIdx.x * 8 + wave;          // wave-uniform column
    const int ncol = lane & 15;                      // N index (0..3 used)
    const int ncl  = (ncol < NCH) ? ncol : (NCH - 1); // clamped N for safe indexing
    const int half = lane >> 4;                      // lane half selects K pair
    const int m    = lane & 15;                      // M (batch) index within half
    const int eBeg = colptr[c];
    const int eEnd = colptr[c + 1];
    const int nnzTot  = rowptr[NN];
    const int strideA = 4 * nnzTot;                  // kernel stride per in-channel tile
    const int a0 = 2 * half, a1 = a0 + 1;

    const float bv = bias[ncl * NN + c];             // junk for N>=4, never stored
    v8f acc0 = {bv, bv, bv, bv, bv, bv, bv, bv};     // batches 0..15
    v8f acc1 = acc0;                                 // batches 16..31

    for (int e = eBeg; e < eEnd; ++e) {
        const int r  = e_r[e];    // wave-uniform -> scalar loads
        const int wb = e_wb[e];
        const int rn = e_rn[e];
        v2f A0, A1, B;
        // A 16x4 f32 layout: VGPR0 = K{0|2}, VGPR1 = K{1|3} split by lane half
        A0.x = x[(size_t)m * XROW + a0 * NN + r];
        A0.y = x[(size_t)m * XROW + a1 * NN + r];
        A1.x = x[(size_t)(m + 16) * XROW + a0 * NN + r];
        A1.y = x[(size_t)(m + 16) * XROW + a1 * NN + r];
        // B 4x16 f32: row K striped across lanes; W[a][bo] = kw[a*4nnz + wb + bo*rn]
        B.x = kw[(size_t)a0 * strideA + wb + ncl * rn];
        B.y = kw[(size_t)a1 * strideA + wb + ncl * rn];
        acc0 = __builtin_amdgcn_wmma_f32_16x16x4_f32(false, A0, false, B,
                                                     (short)0, acc0, false, false);
        acc1 = __builtin_amdgcn_wmma_f32_16x16x4_f32(false, A1, false, B,
                                                     (short)0, acc1, false, false);
    }

    // C/D layout: VGPR i holds M = i + 8*half, N = lane&15
    if (ncol < NCH) {
#pragma unroll
        for (int i = 0; i < 8; ++i) {
            const int mr = i + half * 8;
            out[(size_t)mr * XROW + ncol * NN + c]        = acc0[i];
            out[(size_t)(mr + 16) * XROW + ncol * NN + c] = acc1[i];
        }
    }
}

extern "C" void kernel_launch(void* const* d_in, const int* in_sizes, int n_in,
                              void* d_out, int out_size, void* d_ws, size_t ws_size,
                              hipStream_t stream) {
    (void)in_sizes; (void)n_in; (void)out_size; (void)ws_size;
    const float* inputs    = (const float*)d_in[0];
    const float* adjacency = (const float*)d_in[1];
    const float* kernelArr = (const float*)d_in[2];
    const float* bias      = (const float*)d_in[3];
    float* out = (float*)d_out;

    // workspace layout (ints): row_nnz[2048] | rowptr[2049] | col_nnz[2048] |
    // colptr[2049] | lidx u16[2048*2048] | e_r[65536] | e_wb[65536] | e_rn[65536]
    int* w = (int*)d_ws;
    int* row_nnz = w;
    int* rowptr  = w + 2048;
    int* col_nnz = w + 4104;
    int* colptr  = w + 6152;
    unsigned short* lidx = (unsigned short*)(w + 8208);
    int* e_r  = w + 8208 + (2048 * 2048 / 2);
    int* e_wb = e_r + 65536;
    int* e_rn = e_wb + 65536;

    k_row_scan <<<256, 256, 0, stream>>>(adjacency, row_nnz, lidx);
    k_col_count<<<8,   256, 0, stream>>>(adjacency, col_nnz);
    k_scan2048 <<<1,  1024, 0, stream>>>(row_nnz, rowptr);
    k_scan2048 <<<1,  1024, 0, stream>>>(col_nnz, colptr);
    k_csc_build<<<8,   256, 0, stream>>>(adjacency, rowptr, row_nnz, lidx, colptr,
                                         e_r, e_wb, e_rn);
    k_spmm_wmma<<<256, 256, 0, stream>>>(inputs, kernelArr, bias, out,
                                         colptr, rowptr, e_r, e_wb, e_rn);
}